// GCN_46986942218822
// MI455X (gfx1250) — compile-verified
//
#include <hip/hip_runtime.h>

typedef __attribute__((ext_vector_type(2))) float v2f;
typedef __attribute__((ext_vector_type(8))) float v8f;

#define IN_C  128
#define HID_C 128
#define OUT_C 64

static inline int igrid(long long n, int b) { return (int)((n + b - 1) / b); }
static inline int imin_i(int a, int b) { return a < b ? a : b; }

// ---------------------------------------------------------------------------
// degree / normalization
// ---------------------------------------------------------------------------
__global__ void gcn_deg_init(float* __restrict__ deg, int N) {
    int i = blockIdx.x * blockDim.x + threadIdx.x;
    if (i < N) deg[i] = 1.0f;   // self-loop
}

__global__ void gcn_deg_accum(const long long* __restrict__ dst,
                              float* __restrict__ deg, int E) {
    int e = blockIdx.x * blockDim.x + threadIdx.x;
    if (e < E) unsafeAtomicAdd(&deg[dst[e]], 1.0f);
}

__global__ void gcn_deg_to_dinv(float* __restrict__ deg, int N) {
    int i = blockIdx.x * blockDim.x + threadIdx.x;
    if (i < N) {
        float d = deg[i];
        deg[i] = d > 0.0f ? rsqrtf(d) : 0.0f;
    }
}

__global__ void gcn_zero_f4(float4* __restrict__ p, long long n4) {
    long long i = (long long)blockIdx.x * blockDim.x + threadIdx.x;
    if (i < n4) p[i] = make_float4(0.f, 0.f, 0.f, 0.f);
}

// ---------------------------------------------------------------------------
// fp32 WMMA GEMM:  D[M x (NT*16)] = A[M x 128] * W[128 x (NT*16)]
// One wave per 16-col tile; B operand resident in registers for whole kernel.
// ---------------------------------------------------------------------------
template <int NT>
__global__ __launch_bounds__(NT * 32)
void gcn_gemm_wmma_f32(const float* __restrict__ A,
                       const float* __restrict__ W,
                       float* __restrict__ D,
                       int Mtiles) {
    const int lane  = threadIdx.x & 31;
    const int wave  = threadIdx.x >> 5;       // 0..NT-1 -> column tile
    const int Ncols = NT * 16;
    const int n     = wave * 16 + (lane & 15);
    const int khalf = (lane >> 4) * 2;        // lanes 0-15: K+{0,1}; 16-31: K+{2,3}

    // Preload this wave's B tile (128 x 16) into registers: 32 k-steps of v2f.
    v2f B[32];
#pragma unroll
    for (int s = 0; s < 32; ++s) {
        int k0 = 4 * s + khalf;
        v2f b;
        b.x = W[(k0 + 0) * Ncols + n];
        b.y = W[(k0 + 1) * Ncols + n];
        B[s] = b;
    }

    const int rowin = lane & 15;
    for (int m = blockIdx.x; m < Mtiles; m += gridDim.x) {
        const float* arow = A + (size_t)(m * 16 + rowin) * 128 + khalf;
        v8f acc = {};
#pragma unroll
        for (int s = 0; s < 32; ++s) {
            v2f a = *(const v2f*)(arow + 4 * s);
            acc = __builtin_amdgcn_wmma_f32_16x16x4_f32(
                false, a, false, B[s], (short)0, acc, false, false);
        }
        // C/D layout: VGPR v -> row v (lanes 0-15) / row v+8 (lanes 16-31)
        const int mrow = m * 16 + (lane >> 4) * 8;
#pragma unroll
        for (int v = 0; v < 8; ++v)
            D[(size_t)(mrow + v) * Ncols + n] = acc[v];
    }
}

// scalar tail for M not divisible by 16 (not hit for N=100000, kept for safety)
__global__ void gcn_gemm_tail(const float* __restrict__ A, const float* __restrict__ W,
                              float* __restrict__ D, int M, int m0, int K, int Ncols) {
    int gid = blockIdx.x * blockDim.x + threadIdx.x;
    int rows = M - m0;
    if (gid >= rows * Ncols) return;
    int r = m0 + gid / Ncols, c = gid % Ncols;
    float s = 0.f;
    for (int k = 0; k < K; ++k) s += A[(size_t)r * K + k] * W[(size_t)k * Ncols + c];
    D[(size_t)r * Ncols + c] = s;
}

// ---------------------------------------------------------------------------
// edge propagation: agg[dst] += h[src] * dinv[src]*dinv[dst]   (C = 1<<CSHIFT)
// one thread per (edge, 4-channel chunk)
// ---------------------------------------------------------------------------
template <int CSHIFT>
__global__ void gcn_propagate(const long long* __restrict__ src,
                              const long long* __restrict__ dst,
                              const float* __restrict__ dinv,
                              const float* __restrict__ h,
                              float* __restrict__ agg,
                              long long nwork) {
    long long gid = (long long)blockIdx.x * blockDim.x + threadIdx.x;
    if (gid >= nwork) return;
    const int cs = CSHIFT - 2;
    long long e  = gid >> cs;
    int chunk    = (int)(gid & ((1 << cs) - 1));
    long long s  = src[e], d = dst[e];
    float w = dinv[s] * dinv[d];
    const float4 v = *(const float4*)(h + ((size_t)s << CSHIFT) + chunk * 4);
    float* base = agg + ((size_t)d << CSHIFT) + chunk * 4;
    unsafeAtomicAdd(base + 0, v.x * w);
    unsafeAtomicAdd(base + 1, v.y * w);
    unsafeAtomicAdd(base + 2, v.z * w);
    unsafeAtomicAdd(base + 3, v.w * w);
}

// ---------------------------------------------------------------------------
// layer-1 finalize: h1 = relu(agg1 + hlin*dinv^2 + b1)   (in place over agg1)
// ---------------------------------------------------------------------------
__global__ void gcn_finalize1(float* __restrict__ agg1,
                              const float* __restrict__ hlin,
                              const float* __restrict__ dinv,
                              const float* __restrict__ b1,
                              long long nwork) {   // N * (HID_C/4)
    long long gid = (long long)blockIdx.x * blockDim.x + threadIdx.x;
    if (gid >= nwork) return;
    long long i = gid >> 5;         // HID_C/4 == 32
    int c4 = (int)(gid & 31) * 4;
    float di = dinv[i], w = di * di;
    size_t off = ((size_t)i << 7) + c4;
    float4 a  = *(const float4*)(agg1 + off);
    float4 hl = *(const float4*)(hlin + off);
    float4 bb = *(const float4*)(b1 + c4);
    float4 r;
    r.x = fmaxf(a.x + hl.x * w + bb.x, 0.f);
    r.y = fmaxf(a.y + hl.y * w + bb.y, 0.f);
    r.z = fmaxf(a.z + hl.z * w + bb.z, 0.f);
    r.w = fmaxf(a.w + hl.w * w + bb.w, 0.f);
    *(float4*)(agg1 + off) = r;
}

// ---------------------------------------------------------------------------
// layer-2 finalize + log_softmax: one wave32 per node, 2 channels per lane
// ---------------------------------------------------------------------------
__global__ void gcn_finalize2_lsm(const float* __restrict__ agg2,
                                  const float* __restrict__ t2,
                                  const float* __restrict__ dinv,
                                  const float* __restrict__ b2,
                                  float* __restrict__ out, int N) {
    int node = blockIdx.x * (blockDim.x >> 5) + (threadIdx.x >> 5);
    int lane = threadIdx.x & 31;
    if (node >= N) return;                       // wave-uniform exit
    float di = dinv[node], w = di * di;
    size_t off = (size_t)node * OUT_C + lane * 2;
    float2 a = *(const float2*)(agg2 + off);
    float2 t = *(const float2*)(t2 + off);
    float x0 = a.x + t.x * w + b2[lane * 2];
    float x1 = a.y + t.y * w + b2[lane * 2 + 1];
    float m = fmaxf(x0, x1);
    for (int o = 16; o > 0; o >>= 1) m = fmaxf(m, __shfl_xor(m, o, 32));
    float s = expf(x0 - m) + expf(x1 - m);
    for (int o = 16; o > 0; o >>= 1) s += __shfl_xor(s, o, 32);
    float ls = logf(s);
    float2 r; r.x = x0 - m - ls; r.y = x1 - m - ls;
    *(float2*)(out + off) = r;
}

// ---------------------------------------------------------------------------
extern "C" void kernel_launch(void* const* d_in, const int* in_sizes, int n_in,
                              void* d_out, int out_size, void* d_ws, size_t ws_size,
                              hipStream_t stream) {
    const float*     x   = (const float*)d_in[0];
    const long long* ei  = (const long long*)d_in[1];
    const float*     W1  = (const float*)d_in[2];
    const float*     b1  = (const float*)d_in[3];
    const float*     W2  = (const float*)d_in[4];
    const float*     b2  = (const float*)d_in[5];
    float*           out = (float*)d_out;

    const int       N   = in_sizes[0] / IN_C;
    const long long E   = in_sizes[1] / 2;
    const long long* srcp = ei;
    const long long* dstp = ei + E;

    // workspace layout (floats): deg | hlin(N*128) | agg1(N*128) | agg2(N*64) | t2(N*64)
    float* ws   = (float*)d_ws;
    size_t Np   = ((size_t)N + 63) & ~(size_t)63;
    float* deg  = ws;                      // becomes dinv in place
    float* hlin = ws + Np;
    float* agg1 = hlin + (size_t)N * HID_C;   // becomes h1 in place
    float* agg2 = agg1 + (size_t)N * HID_C;
    float* t2   = agg2 + (size_t)N * OUT_C;

    // 1. degrees -> dinv
    gcn_deg_init<<<igrid(N, 256), 256, 0, stream>>>(deg, N);
    gcn_deg_accum<<<igrid(E, 256), 256, 0, stream>>>(dstp, deg, (int)E);
    gcn_deg_to_dinv<<<igrid(N, 256), 256, 0, stream>>>(deg, N);

    // 2. zero accumulators (agg1 and agg2 are contiguous: N*(128+64) floats)
    long long nz4 = (long long)N * (HID_C + OUT_C) / 4;
    gcn_zero_f4<<<igrid(nz4, 256), 256, 0, stream>>>((float4*)agg1, nz4);

    // 3. layer-1 GEMM: hlin = x @ W1
    int Mt = N / 16, Mtail = N - Mt * 16;
    if (Mt > 0)
        gcn_gemm_wmma_f32<HID_C / 16><<<imin_i(Mt, 4096), (HID_C / 16) * 32, 0, stream>>>(
            x, W1, hlin, Mt);
    if (Mtail > 0)
        gcn_gemm_tail<<<igrid((long long)Mtail * HID_C, 256), 256, 0, stream>>>(
            x, W1, hlin, N, Mt * 16, IN_C, HID_C);

    // 4. layer-1 propagation (real edges; self-loop folded into finalize)
    long long nw1 = E * (HID_C / 4);
    gcn_propagate<7><<<igrid(nw1, 256), 256, 0, stream>>>(srcp, dstp, deg, hlin, agg1, nw1);

    // 5. finalize layer 1: h1 = relu(agg1 + hlin*dinv^2 + b1)
    gcn_finalize1<<<igrid((long long)N * (HID_C / 4), 256), 256, 0, stream>>>(
        agg1, hlin, deg, b1, (long long)N * (HID_C / 4));

    // 6. layer-2 GEMM: t2 = h1 @ W2
    if (Mt > 0)
        gcn_gemm_wmma_f32<OUT_C / 16><<<imin_i(Mt, 4096), (OUT_C / 16) * 32, 0, stream>>>(
            agg1, W2, t2, Mt);
    if (Mtail > 0)
        gcn_gemm_tail<<<igrid((long long)Mtail * OUT_C, 256), 256, 0, stream>>>(
            agg1, W2, t2, N, Mt * 16, HID_C, OUT_C);

    // 7. layer-2 propagation
    long long nw2 = E * (OUT_C / 4);
    gcn_propagate<6><<<igrid(nw2, 256), 256, 0, stream>>>(srcp, dstp, deg, t2, agg2, nw2);

    // 8. finalize + log_softmax
    gcn_finalize2_lsm<<<igrid(N, 8), 256, 0, stream>>>(agg2, t2, deg, b2, out, N);
}